// QueryAndGroup_16947940950603
// MI455X (gfx1250) — compile-verified
//
#include <hip/hip_runtime.h>

// Problem constants from the reference
#define RADIUS2   0.01f       // 0.1^2
#define NSAMPLE   32
#define BQ        8
#define NPTS      8192
#define SQ        2048
#define CF        128
#define CHANNELS  134         // 3 (centered) + 3 (centered again) + 128 (features)

typedef float v2f __attribute__((ext_vector_type(2)));
typedef float v8f __attribute__((ext_vector_type(8)));

#define WAVES_PER_BLOCK 4     // 128 threads = 4 wave32s
#define NTILES          (NPTS / 16)   // 512

__global__ __launch_bounds__(WAVES_PER_BLOCK * 32)
void QueryAndGroup_16947940950603_kernel(const float* __restrict__ xyz,      // (B, N, 3)
                                         const float* __restrict__ new_xyz,  // (B, S, 3)
                                         const float* __restrict__ feats,    // (B, C, N)
                                         float* __restrict__ out)            // (B, 134, S, 32)
{
    __shared__ int s_cnt[WAVES_PER_BLOCK][16];
    __shared__ int s_idx[WAVES_PER_BLOCK][16][NSAMPLE];

    const int w    = threadIdx.x >> 5;
    const int lane = threadIdx.x & 31;
    const int m    = lane & 15;    // column within tile / query row id
    const int half = lane >> 4;    // which half-wave (selects K pair for A/B)

    // One wave handles 16 query points. 1024 tiles total.
    const int tile          = blockIdx.x * WAVES_PER_BLOCK + w;
    const int tilesPerBatch = SQ / 16;                 // 128
    const int b             = tile / tilesPerBatch;
    const int s0            = (tile % tilesPerBatch) * 16;

    // ---- Build A matrix (16x4 f32): row m = (-2qx, -2qy, -2qz, 1) ----
    // 32-bit 16x4 A layout: lanes 0-15 hold K=0 (A.x), K=1 (A.y); lanes 16-31 hold K=2, K=3.
    const float* qp = new_xyz + ((size_t)b * SQ + (s0 + m)) * 3;
    const float qx = qp[0], qy = qp[1], qz = qp[2];
    const float qn2 = qx * qx + qy * qy + qz * qz;

    v2f A;
    A.x = half ? (-2.0f * qz) : (-2.0f * qx);
    A.y = half ? 1.0f         : (-2.0f * qy);

    // Per-row compare threshold: d_wmma[r] = |p|^2 - 2 q.p, hit iff <= r^2 - |q_row|^2
    float thresh[8];
#pragma unroll
    for (int r = 0; r < 8; ++r)
        thresh[r] = RADIUS2 - __shfl(qn2, r + 8 * half, 32);

    const float* xb = xyz + (size_t)b * NPTS * 3;

    // Per-lane shadow counters (uniform within each half-wave): row r + 8*half
    unsigned cnt[8];
#pragma unroll
    for (int r = 0; r < 8; ++r) cnt[r] = 0;

    // Software pipeline: preload tile 0
    float px = xb[(size_t)m * 3 + 0];
    float py = xb[(size_t)m * 3 + 1];
    float pz = xb[(size_t)m * 3 + 2];

    for (int t = 0; t < NTILES; ++t) {
        // Issue next tile's load early (clamped to stay in-bounds on last iter)
        const int tn = (t + 1 < NTILES) ? (t + 1) : (NTILES - 1);
        const float* pnx = xb + ((size_t)tn * 16 + m) * 3;
        const float nx = pnx[0], ny = pnx[1], nz = pnx[2];
        __builtin_prefetch(xb + ((size_t)tn * 16 + 256 + m) * 3, 0, 1);

        const float pn2 = px * px + py * py + pz * pz;

        // B matrix (4x16 f32): col = (px, py, pz, |p|^2)
        v2f B;
        B.x = half ? pz  : px;
        B.y = half ? pn2 : py;

        v8f c = {};  // inline-0 accumulator
        v8f d = __builtin_amdgcn_wmma_f32_16x16x4_f32(
            false, A, false, B, (short)0, c, false, false);

        // Ballots first (cheap v_cmp -> SGPR), then skip all bookkeeping when no hits.
        unsigned bal[8];
        unsigned any = 0;
#pragma unroll
        for (int r = 0; r < 8; ++r) {
            bal[r] = (unsigned)__ballot(d[r] <= thresh[r]);
            any |= bal[r];
        }

        if (any) {
            const int n = t * 16 + m;
#pragma unroll
            for (int r = 0; r < 8; ++r) {
                if (bal[r]) {                      // wave-uniform branch
                    const unsigned hm = (bal[r] >> (16 * half)) & 0xFFFFu;
                    const int pos = (int)cnt[r] + __popc(hm & ((1u << m) - 1u));
                    if (((hm >> m) & 1u) && pos < NSAMPLE)
                        s_idx[w][r + 8 * half][pos] = n;
                    cnt[r] += __popc(hm);
                }
            }
        }

        px = nx; py = ny; pz = nz;
    }

    // Spill final counts (lane 0 -> rows 0-7, lane 16 -> rows 8-15)
    if (m == 0) {
#pragma unroll
        for (int r = 0; r < 8; ++r)
            s_cnt[w][r + 8 * half] = (int)cnt[r];
    }

    // ---- Padding per reference: pad with first hit, or 0 if no hits ----
#pragma unroll 1
    for (int row = 0; row < 16; ++row) {
        const int c0    = s_cnt[w][row];
        const int first = (c0 > 0) ? s_idx[w][row][0] : 0;
        const int cc    = c0 < NSAMPLE ? c0 : NSAMPLE;
        if (lane >= cc) s_idx[w][row][lane] = first;
    }

    // ---- Grouping: lane == sample -> fully coalesced 128B stores ----
    float* outb = out + (size_t)b * CHANNELS * SQ * NSAMPLE;
    const float* fbb = feats + (size_t)b * CF * NPTS;
    const size_t chs = (size_t)SQ * NSAMPLE;

#pragma unroll 1
    for (int row = 0; row < 16; ++row) {
        const int s  = s0 + row;
        const int id = s_idx[w][row][lane];

        const float* qq = new_xyz + ((size_t)b * SQ + s) * 3;
        const float* pp = xb + (size_t)id * 3;
        const float cx = pp[0] - qq[0];
        const float cy = pp[1] - qq[1];
        const float cz = pp[2] - qq[2];

        const size_t base = (size_t)s * NSAMPLE + lane;
        outb[0 * chs + base] = cx;
        outb[1 * chs + base] = cy;
        outb[2 * chs + base] = cz;
        outb[3 * chs + base] = cx;
        outb[4 * chs + base] = cy;
        outb[5 * chs + base] = cz;

        const float* fptr = fbb + id;
#pragma unroll 4
        for (int cch = 0; cch < CF; ++cch)
            outb[(size_t)(6 + cch) * chs + base] = fptr[(size_t)cch * NPTS];
    }
}

extern "C" void kernel_launch(void* const* d_in, const int* in_sizes, int n_in,
                              void* d_out, int out_size, void* d_ws, size_t ws_size,
                              hipStream_t stream) {
    (void)in_sizes; (void)n_in; (void)d_ws; (void)ws_size; (void)out_size;
    const float* xyz     = (const float*)d_in[0];
    const float* new_xyz = (const float*)d_in[1];
    const float* feats   = (const float*)d_in[2];
    float* out           = (float*)d_out;

    const int totalTiles = BQ * SQ / 16;                    // 1024 waves
    const int blocks     = totalTiles / WAVES_PER_BLOCK;    // 256 blocks x 128 threads
    QueryAndGroup_16947940950603_kernel<<<blocks, WAVES_PER_BLOCK * 32, 0, stream>>>(
        xyz, new_xyz, feats, out);
}